// QueryLayer_35966056136793
// MI455X (gfx1250) — compile-verified
//
#include <hip/hip_runtime.h>
#include <hip/hip_bf16.h>

// ---------------------------------------------------------------------------
// QueryLayer on MI455X (gfx1250, wave32)
//
// Stage 1: per-point 27-neighbor gather + coordinate-attention (memory bound,
//          ~346 MB of coalesced 128B gathers -> ~15us at 23.3 TB/s).
//          Algebra: sim-sum normalization cancels against the feature-sum
//          normalization, so feat[c] = acc[c] / sum_c acc[c].
// Stage 2: [N,32] x [32,128] + bias via V_WMMA_F32_16X16X4_F32 (full f32
//          matrix path; GEMM is only 0.82 GFLOP so precision downcast buys
//          nothing -- keep it exact). Epilogue uses a wave-uniform in-bounds
//          fast path so the 8 D-row stores are unpredicated global_store_b32
//          off one base address with immediate offsets.
// ---------------------------------------------------------------------------

typedef __attribute__((ext_vector_type(2))) float v2f;
typedef __attribute__((ext_vector_type(8))) float v8f;

#define VOL_X 304
#define VOL_Y 304
#define VOL_Z 128
#define VOL_C 32
#define OUT_DIM 128

// --------------------------- Stage 1: gather ------------------------------
// One wave per point; lane = channel (VOL_C == 32 == wave32). Each neighbor
// fetch is one contiguous 128-byte line across the wave.
__global__ void qlayer_gather_attn(const float* __restrict__ pts,
                                   const float* __restrict__ vol,
                                   float* __restrict__ feat,
                                   int N) {
    const int lane  = threadIdx.x & 31;
    const int point = (blockIdx.x * blockDim.x + threadIdx.x) >> 5;
    if (point >= N) return;  // wave-uniform exit

    const float px = pts[point * 3 + 0];
    const float py = pts[point * 3 + 1];
    const float pz = pts[point * 3 + 2];

    // _transform
    const float qx = px + 24.0f;
    const float qy = py + 24.0f;
    const float qz = (pz - 425.0f) * (128.0f / 480.0f);

    // center = (floor(q)+ceil(q))/2 ; lo = floor(center - 1.5)
    const float cx = (floorf(qx) + ceilf(qx)) * 0.5f;
    const float cy = (floorf(qy) + ceilf(qy)) * 0.5f;
    const float cz = (floorf(qz) + ceilf(qz)) * 0.5f;
    const int lx = (int)floorf(cx - 1.5f);
    const int ly = (int)floorf(cy - 1.5f);
    const int lz = (int)floorf(cz - 1.5f);

    float acc = 0.0f;
#pragma unroll
    for (int k = 0; k < 27; ++k) {
        const int x = lx + (k / 9);
        const int y = ly + ((k / 3) % 3);
        const int z = lz + (k % 3);
        // attention logit = key . q  (key is the integer coordinate)
        const float w = (float)x * qx + (float)y * qy + (float)z * qz;
        const size_t idx =
            ((((size_t)x * VOL_Y + (size_t)y) * VOL_Z + (size_t)z) * VOL_C) +
            (size_t)lane;
        acc = fmaf(w, vol[idx], acc);
    }

    // sum over channels (32 lanes), then normalize: sim_sum cancels out.
    float tot = acc;
#pragma unroll
    for (int m = 16; m >= 1; m >>= 1) tot += __shfl_xor(tot, m, 32);

    feat[(size_t)point * VOL_C + lane] = acc / tot;
}

// --------------------------- Stage 2: WMMA GEMM ---------------------------
// out[n, o] = sum_c feat[n,c] * W[o,c] + b[o]
// A = feat (16x4 slices), B = W^T (4x16 slices), C/D = 16x16 f32.
// A layout (ISA 7.12.2): lanes 0-15 hold K=0,1 (VGPR0,1); lanes 16-31 hold
// K=2,3. B/C/D: row striped across lanes within a VGPR; D VGPR j = row j
// (lanes 0-15) / row j+8 (lanes 16-31).
__global__ void qlayer_wmma_gemm(const float* __restrict__ feat,
                                 const float* __restrict__ W,
                                 const float* __restrict__ bias,
                                 float* __restrict__ out,
                                 int N) {
    const int lane     = threadIdx.x & 31;
    const int waveGbl  = (blockIdx.x * blockDim.x + threadIdx.x) >> 5;
    const int nRowTile = (N + 15) >> 4;

    const int rowTile = waveGbl >> 3;   // 8 column tiles (OUT_DIM/16)
    const int colTile = waveGbl & 7;
    if (rowTile >= nRowTile) return;    // wave-uniform: EXEC all-ones below

    const int row0 = rowTile << 4;
    const int col0 = colTile << 4;
    const int half = lane >> 4;         // 0: lanes 0-15, 1: lanes 16-31
    const int l    = lane & 15;

    // Clamp A row for a possible ragged tail (N=100000 is a multiple of 16,
    // so this is a no-op there; keeps EXEC uniform either way).
    int arow = row0 + l;
    if (arow > N - 1) arow = N - 1;
    const float* aptr = feat + (size_t)arow * VOL_C;
    const float* bptr = W + (size_t)(col0 + l) * VOL_C;  // W[o, c], o = col0+l

    v8f c = {};
#pragma unroll
    for (int k0 = 0; k0 < VOL_C; k0 += 4) {
        v2f a, bm;
        const int ka = k0 + half * 2;
        a.x  = aptr[ka + 0];     // K = ka   for this lane's row
        a.y  = aptr[ka + 1];     // K = ka+1
        bm.x = bptr[ka + 0];     // B[ka,   col0+l] = W[col0+l, ka]
        bm.y = bptr[ka + 1];     // B[ka+1, col0+l]
        // 8 args: (neg_a, A, neg_b, B, c_mod, C, reuse_a, reuse_b)
        c = __builtin_amdgcn_wmma_f32_16x16x4_f32(
                false, a, false, bm, (short)0, c, false, false);
    }

    const float bv = bias[col0 + l];

    // D layout: VGPR j -> row row0+j (lanes 0-15) / row0+j+8 (lanes 16-31).
    // One base pointer; per-row offsets are compile-time immediates.
    float* op = out + (size_t)(row0 + half * 8) * OUT_DIM + (col0 + l);

    if (row0 + 16 <= N) {
        // Wave-uniform in-bounds fast path: 8 unpredicated stores, imm offsets.
#pragma unroll
        for (int j = 0; j < 8; ++j) {
            op[(size_t)j * OUT_DIM] = c[j] + bv;
        }
    } else {
        // Ragged tail (never taken for N % 16 == 0).
#pragma unroll
        for (int j = 0; j < 8; ++j) {
            const int r = row0 + j + half * 8;
            if (r < N) op[(size_t)j * OUT_DIM] = c[j] + bv;
        }
    }
}

// ------------------------------- launcher ---------------------------------
extern "C" void kernel_launch(void* const* d_in, const int* in_sizes, int n_in,
                              void* d_out, int out_size, void* d_ws, size_t ws_size,
                              hipStream_t stream) {
    const float* pts  = (const float*)d_in[0];  // [N,3]
    const float* vol  = (const float*)d_in[1];  // [304,304,128,32]
    const float* W    = (const float*)d_in[2];  // [128,32]
    const float* bias = (const float*)d_in[3];  // [128]
    float* out = (float*)d_out;                 // [N,128]
    const int N = in_sizes[0] / 3;

    float* feat = (float*)d_ws;                 // [N,32] f32 scratch

    // Stage 1: one wave per point, 8 waves per 256-thread block.
    {
        const int threads = 256;
        const int waves   = N;
        const int blocks  = (waves * 32 + threads - 1) / threads;
        qlayer_gather_attn<<<blocks, threads, 0, stream>>>(pts, vol, feat, N);
    }

    // Stage 2: one wave per 16x16 output tile; (N/16) x 8 tiles.
    {
        const int threads   = 256;               // 8 waves/block
        const int nRowTile  = (N + 15) / 16;
        const int waves     = nRowTile * 8;
        const int blocks    = (waves + 7) / 8;
        qlayer_wmma_gemm<<<blocks, threads, 0, stream>>>(feat, W, bias, out, N);
    }
}